// EMA_56599079027011
// MI455X (gfx1250) — compile-verified
//
#include <hip/hip_runtime.h>
#include <math.h>

// Problem constants (B=32, C=256, H=W=64, GROUPS=32 -> BG=1024, CG=8)
#define BGN   1024
#define CGN   8
#define HH    64
#define WW    64
#define HP    66                 // padded row (halo for 3x3 SAME conv)
#define CH_STR (HP*HP)           // 4356 floats per padded channel plane
#define TILE_ELEMS (CGN*HH*WW)   // 32768 floats per bg slice
#define EPSV  1e-5f

typedef float v2f __attribute__((ext_vector_type(2)));
typedef float v8f __attribute__((ext_vector_type(8)));
typedef unsigned int u32x4 __attribute__((ext_vector_type(4)));
typedef int i32x4 __attribute__((ext_vector_type(4)));
typedef int i32x8 __attribute__((ext_vector_type(8)));

// LDS layout (float offsets)
#define OFF_GXP 0        // 8 x 66 x 66 padded input tile   (34848)
#define OFF_X2  34848    // conv output 8 x 4096            (32768)
#define OFF_W3  67616    // conv weights (576)
#define OFF_SR  68192    // sigmoid row gates 8x64 (512)
#define OFF_SC  68704    // sigmoid col gates 8x64 (512)
#define OFF_W1  69216    // (64)
#define OFF_B1  69280    // (8)
#define OFF_B3  69288    // (8)
#define OFF_GNW 69296    // (8)
#define OFF_GNB 69304    // (8)
#define OFF_MU  69312    // (8)
#define OFF_INV 69320    // rsqrt(var+eps)*gn_w (8)
#define OFF_X2S 69328    // conv-output channel sums (8)
#define OFF_X11 69336    // softmax weights (8)
#define OFF_X21 69344    // softmax weights (8)
#define OFF_XH  69352    // row means 8x64 (512)
#define OFF_XW  69864    // col means 8x64 (512)
#define LDS_FLOATS 70376 // 281,504 bytes <= 320 KB WGP LDS

extern "C" __global__ void __launch_bounds__(256, 1)
ema_fused_kernel(const float* __restrict__ x,  const float* __restrict__ w1,
                 const float* __restrict__ b1, const float* __restrict__ w3,
                 const float* __restrict__ b3, const float* __restrict__ gnw,
                 const float* __restrict__ gnb, float* __restrict__ out)
{
    extern __shared__ float s[];
    const int tid  = threadIdx.x;
    const int bg   = blockIdx.x;
    const int lane = tid & 31;
    const int wave = tid >> 5;
    const long gbaseE = (long)bg * TILE_ELEMS;

    // ---------------- Phase 0: params to LDS, zero halo ----------------
    for (int i = tid; i < 576; i += 256) s[OFF_W3 + i] = w3[i];
    if (tid < 64) s[OFF_W1 + tid] = w1[tid];
    if (tid < 8) {
        s[OFF_B1 + tid] = b1[tid];  s[OFF_B3 + tid]  = b3[tid];
        s[OFF_GNW + tid] = gnw[tid]; s[OFF_GNB + tid] = gnb[tid];
        s[OFF_X2S + tid] = 0.f;
    }
    for (int i = tid; i < 8 * 260; i += 256) {   // halo cells only
        int c = i / 260, r = i % 260, y, xx;
        if (r < 66)       { y = 0;        xx = r; }
        else if (r < 132) { y = 65;       xx = r - 66; }
        else if (r < 196) { y = r - 131;  xx = 0; }
        else              { y = r - 195;  xx = 65; }
        s[OFF_GXP + c * CH_STR + y * HP + xx] = 0.f;
    }

    // ---------------- Phase 1: TDM async load of the tile into padded LDS ---
#if __has_builtin(__builtin_amdgcn_tensor_load_to_lds)
    if (tid < 32) {  // one wave issues 8 tensor DMAs (one per channel plane)
        unsigned lds_base = (unsigned)(unsigned long long)(&s[OFF_GXP]);
        unsigned long long gb = (unsigned long long)x + (unsigned long long)gbaseE * 4ull;
#pragma unroll
        for (int c = 0; c < 8; ++c) {
            unsigned long long ga = gb + (unsigned long long)c * 4096ull * 4ull;
            unsigned la = lds_base + (unsigned)((c * CH_STR + HP + 1) * 4);
            u32x4 g0; i32x8 g1; i32x4 g2 = {0,0,0,0}, g3 = {0,0,0,0};
            g0[0] = 1u;                                    // count=1 valid descriptor
            g0[1] = la;                                    // lds_addr (bytes)
            g0[2] = (unsigned)(ga & 0xffffffffu);          // global_addr[31:0]
            g0[3] = (unsigned)((ga >> 32) & 0x1ffffffu) | (2u << 30); // ga[56:32] | type=2
            // data_size=4B, pad_enable, pad_interval=64DW(enc 5), pad_amount=2DW(enc 1)
            g1[0] = (int)((2u << 16) | (1u << 20) | (5u << 22) | (1u << 25));
            g1[1] = (int)(64u << 16);   // tensor_dim0 = 64
            g1[2] = (int)(64u << 16);   // tensor_dim1 = 64
            g1[3] = (int)(64u << 16);   // tile_dim0   = 64
            g1[4] = 64;                 // tile_dim1   = 64
            g1[5] = 64;                 // tensor_dim0_stride = 64
            g1[6] = 0; g1[7] = 0;
#if __clang_major__ >= 23
            i32x8 g4 = {0,0,0,0,0,0,0,0};
            __builtin_amdgcn_tensor_load_to_lds(g0, g1, g2, g3, g4, 0);
#else
            __builtin_amdgcn_tensor_load_to_lds(g0, g1, g2, g3, 0);
#endif
        }
        __builtin_amdgcn_s_wait_tensorcnt(0);
    }
#else
    for (int q = tid; q < 8192; q += 256) {      // fallback: manual copy
        int c = q >> 10, rem = q & 1023, h = rem >> 4, w4 = (rem & 15) * 4;
        float4 v = ((const float4*)(x + gbaseE))[q];
        float* d = &s[OFF_GXP + c * CH_STR + (h + 1) * HP + 1 + w4];
        d[0] = v.x; d[1] = v.y; d[2] = v.z; d[3] = v.w;
    }
#endif
    __syncthreads();

    // ---------------- Phase 2: row/col means -> gates -------------------
    for (int i = tid; i < 512; i += 256) {       // x_h: mean over w
        int c = i >> 6, h = i & 63;
        const float* p = &s[OFF_GXP + c * CH_STR + (h + 1) * HP + 1];
        float acc = 0.f;
#pragma unroll 8
        for (int w = 0; w < 64; ++w) acc += p[w];
        s[OFF_XH + i] = acc * (1.f / 64.f);
    }
    for (int i = tid; i < 512; i += 256) {       // x_w: mean over h
        int c = i >> 6, w = i & 63;
        const float* p = &s[OFF_GXP + c * CH_STR + HP + 1 + w];
        float acc = 0.f;
#pragma unroll 8
        for (int h = 0; h < 64; ++h) acc += p[h * HP];
        s[OFF_XW + i] = acc * (1.f / 64.f);
    }
    __syncthreads();
    for (int i = tid; i < 1024; i += 256) {      // hw = w1@[xh;xw]+b1 -> sigmoid
        int o = i >> 7, k = i & 127;
        float acc = s[OFF_B1 + o];
#pragma unroll
        for (int c = 0; c < 8; ++c)
            acc += s[OFF_W1 + o * 8 + c] *
                   ((k < 64) ? s[OFF_XH + c * 64 + k] : s[OFF_XW + c * 64 + (k - 64)]);
        float g = 1.f / (1.f + expf(-acc));
        if (k < 64) s[OFF_SR + o * 64 + k] = g;
        else        s[OFF_SC + o * 64 + (k - 64)] = g;
    }
    __syncthreads();

    // ---------------- Phase 3: per-channel mean/var of gated ------------
    {
        const int c = wave;                       // 8 waves <-> 8 channels
        float sm = 0.f, ss = 0.f;
        for (int l = lane; l < 4096; l += 32) {
            int h = l >> 6, w = l & 63;
            float g = s[OFF_GXP + c * CH_STR + (h + 1) * HP + (w + 1)] *
                      s[OFF_SR + c * 64 + h] * s[OFF_SC + c * 64 + w];
            sm += g; ss = fmaf(g, g, ss);
        }
#pragma unroll
        for (int o = 16; o > 0; o >>= 1) { sm += __shfl_xor(sm, o, 32); ss += __shfl_xor(ss, o, 32); }
        if (lane == 0) {
            float mu = sm * (1.f / 4096.f);
            float var = ss * (1.f / 4096.f) - mu * mu;
            s[OFF_MU + c]  = mu;
            s[OFF_INV + c] = rsqrtf(var + EPSV) * s[OFF_GNW + c];
        }
    }
    __syncthreads();

    // ---------------- Phase 4: 3x3 conv via v_wmma_f32_16x16x4_f32 ------
    // K-order: K = kh3*24 + kw3*8 + ci (72 = 18 chunks of 4).
    // A(16x4): lane m holds co=m; K = vgpr + 2*(lane>=16). B(4x16): lane n holds
    // spatial col; same K striping. D: VGPR r = out-channel r on lanes 0-15.
    {
        const int m = lane & 15;
        const int half = lane >> 4;
        float aF[36]; int dOf[36];
#pragma unroll
        for (int kc = 0; kc < 18; ++kc)
#pragma unroll
            for (int j = 0; j < 2; ++j) {
                int K = kc * 4 + j + 2 * half;
                int ci = K & 7, t = K >> 3, kw3 = t % 3, kh3 = t / 3;
                aF[kc * 2 + j] = (m < 8) ? s[OFF_W3 + m * 72 + ci * 9 + kh3 * 3 + kw3] : 0.f;
                dOf[kc * 2 + j] = ci * CH_STR + kh3 * HP + kw3;
            }
        float cs[8];
#pragma unroll
        for (int c = 0; c < 8; ++c) cs[c] = 0.f;

        for (int t8 = 0; t8 < 32; ++t8) {
            int tile = t8 * 8 + wave;            // 256 tiles of 16 positions
            int l = tile * 16 + m;
            int h = l >> 6, w = l & 63;
            int bb = h * HP + w;                 // halo-free base (padded layout)
            v8f acc = {0.f,0.f,0.f,0.f,0.f,0.f,0.f,0.f};
#pragma unroll
            for (int kc = 0; kc < 18; ++kc) {
                v2f A;  A[0]  = aF[kc * 2]; A[1]  = aF[kc * 2 + 1];
                v2f Bv; Bv[0] = s[OFF_GXP + bb + dOf[kc * 2]];
                        Bv[1] = s[OFF_GXP + bb + dOf[kc * 2 + 1]];
                acc = __builtin_amdgcn_wmma_f32_16x16x4_f32(
                          false, A, false, Bv, (short)0, acc, false, false);
            }
            if (lane < 16) {
#pragma unroll
                for (int c = 0; c < 8; ++c) {
                    float v = acc[c] + s[OFF_B3 + c];
                    s[OFF_X2 + c * 4096 + l] = v;
                    cs[c] += v;
                }
            }
        }
#pragma unroll
        for (int c = 0; c < 8; ++c) {
            float v = cs[c];
#pragma unroll
            for (int o = 8; o > 0; o >>= 1) v += __shfl_xor(v, o, 32);
            if (lane == 0) atomicAdd(&s[OFF_X2S + c], v);
        }
    }
    __syncthreads();

    // ---------------- Phase 5: the two 8-wide softmaxes -----------------
    if (tid == 0) {
        // mean(x1) over space == gn_b exactly (normalized mean is 0)
        float m1 = -1e30f, m2 = -1e30f;
        for (int c = 0; c < 8; ++c) {
            m1 = fmaxf(m1, s[OFF_GNB + c]);
            m2 = fmaxf(m2, s[OFF_X2S + c] * (1.f / 4096.f));
        }
        float s1 = 0.f, s2 = 0.f; float e1[8], e2[8];
        for (int c = 0; c < 8; ++c) {
            e1[c] = expf(s[OFF_GNB + c] - m1);                    s1 += e1[c];
            e2[c] = expf(s[OFF_X2S + c] * (1.f / 4096.f) - m2);   s2 += e2[c];
        }
        for (int c = 0; c < 8; ++c) {
            s[OFF_X11 + c] = e1[c] / s1;
            s[OFF_X21 + c] = e2[c] / s2;
        }
    }
    __syncthreads();

    // ---------------- Phase 6: y -> sigmoid -> out = gx * weight --------
    for (int p = 0; p < 4; ++p) {
        int l0 = p * 1024 + tid * 4;             // 4 consecutive positions
        int h = l0 >> 6, w0 = l0 & 63;
        float wt[4];
#pragma unroll
        for (int sI = 0; sI < 4; ++sI) {
            int w = w0 + sI;
            float y = 0.f;
#pragma unroll
            for (int c = 0; c < 8; ++c) {
                float g = s[OFF_GXP + c * CH_STR + (h + 1) * HP + (w + 1)];
                float gated = g * s[OFF_SR + c * 64 + h] * s[OFF_SC + c * 64 + w];
                float x1v = (gated - s[OFF_MU + c]) * s[OFF_INV + c] + s[OFF_GNB + c];
                float x2v = s[OFF_X2 + c * 4096 + l0 + sI];
                y += s[OFF_X11 + c] * x2v + s[OFF_X21 + c] * x1v;
            }
            wt[sI] = 1.f / (1.f + expf(-y));
        }
#pragma unroll
        for (int c = 0; c < 8; ++c) {
            const float* gp = &s[OFF_GXP + c * CH_STR + (h + 1) * HP + (w0 + 1)];
            float4 vo;
            vo.x = gp[0] * wt[0]; vo.y = gp[1] * wt[1];
            vo.z = gp[2] * wt[2]; vo.w = gp[3] * wt[3];
            *(float4*)(out + gbaseE + c * 4096 + l0) = vo;
        }
    }
}

extern "C" void kernel_launch(void* const* d_in, const int* in_sizes, int n_in,
                              void* d_out, int out_size, void* d_ws, size_t ws_size,
                              hipStream_t stream) {
    (void)in_sizes; (void)n_in; (void)out_size; (void)d_ws; (void)ws_size;
    const float* x  = (const float*)d_in[0];
    const float* w1 = (const float*)d_in[1];
    const float* b1 = (const float*)d_in[2];
    const float* w3 = (const float*)d_in[3];
    const float* b3 = (const float*)d_in[4];
    const float* gw = (const float*)d_in[5];
    const float* gb = (const float*)d_in[6];
    float* out = (float*)d_out;

    size_t shmem = (size_t)LDS_FLOATS * sizeof(float);   // ~275 KB (<=320 KB WGP LDS)
    (void)hipFuncSetAttribute((const void*)ema_fused_kernel,
                              hipFuncAttributeMaxDynamicSharedMemorySize, (int)shmem);
    ema_fused_kernel<<<BGN, 256, shmem, stream>>>(x, w1, b1, w3, b3, gw, gb, out);
}